// BaseHomogenousModel_79542794322224
// MI455X (gfx1250) — compile-verified
//
#include <hip/hip_runtime.h>
#include <hip/hip_bf16.h>

#define Nn      100000
#define Ee      1600000
#define EfE     (Ee + Nn)        // edges + self loops
#define FIN     128
#define HIDd    64
#define HEADS   2
#define NL      3
#define EDGED   32
#define BG      1000
#define DT0     3
#define DTS     22
#define SHIN    150              // 128 + 22
#define TASKH   64
#define MAXO    9
#define NSLOPE  0.2f

#define LDSPAD  132              // padded A-tile row stride (floats); avoids 16-way bank conflicts

typedef __attribute__((ext_vector_type(2))) float v2f;
typedef __attribute__((ext_vector_type(8))) float v8f;

// ---------------- utility ----------------
__global__ void k_zero_f(float* p, long n) {
    long i = blockIdx.x * (long)blockDim.x + threadIdx.x;
    if (i < n) p[i] = 0.f;
}

// monotone float <-> u32 key for atomic segment-max
__device__ __forceinline__ unsigned enc_f(float x) {
    unsigned u = __float_as_uint(x);
    return (u >> 31) ? ~u : (u | 0x80000000u);
}
__device__ __forceinline__ float dec_f(unsigned k) {
    return (k >> 31) ? __uint_as_float(k ^ 0x80000000u) : __uint_as_float(~k);
}

// ---------------- degree + mean self-loop edge_attr ----------------
__global__ void k_deg_ealoop(const int* __restrict__ ei, const float* __restrict__ ea,
                             float* __restrict__ deg, float* __restrict__ eal) {
    long t = blockIdx.x * (long)blockDim.x + threadIdx.x;
    if (t >= (long)Ee * EDGED) return;
    long e = t >> 5; int d = (int)(t & 31);
    int dst = ei[Ee + e];
    atomicAdd(&eal[(long)dst * EDGED + d], ea[e * EDGED + d]);
    if (d == 0) atomicAdd(&deg[dst], 1.0f);
}
__global__ void k_ealoop_div(float* __restrict__ eal, const float* __restrict__ deg) {
    long t = blockIdx.x * (long)blockDim.x + threadIdx.x;
    if (t >= (long)Nn * EDGED) return;
    eal[t] /= fmaxf(deg[t >> 5], 1.0f);
}

// ---------------- node0 = searchsorted(batch, g); dtype_vec ----------------
__global__ void k_node0(const int* __restrict__ batch, const float* __restrict__ x,
                        int* __restrict__ node0, float* __restrict__ dtv) {
    int g = blockIdx.x * blockDim.x + threadIdx.x;
    if (g >= BG) return;
    int lo = 0, hi = Nn;
    while (lo < hi) { int mid = (lo + hi) >> 1; if (batch[mid] < g) lo = mid + 1; else hi = mid; }
    node0[g] = lo;
    for (int c = 0; c < DTS; c++) dtv[g * DTS + c] = x[(long)lo * FIN + DT0 + c];
}

// ---------------- ve[d][h] = sum_j eW[d, h*64+j] * a_e[h,j]  (32x2) ----------------
__global__ void k_ve(const float* __restrict__ eW, const float* __restrict__ ae, float* __restrict__ ve) {
    int t = threadIdx.x;
    if (t >= EDGED * HEADS) return;
    int d = t >> 1, h = t & 1;
    float s = 0.f;
    for (int j = 0; j < HIDd; j++) s += eW[d * (HEADS * HIDd) + h * HIDd + j] * ae[h * HIDd + j];
    ve[d * HEADS + h] = s;
}

// =====================================================================
// Specialized node-projection GEMM: C[M x 128] = A[M x 128] * B[128 x 128]
// 8 waves / block; block covers 128 rows; wave w covers rows w*16..w*16+15.
// A tile staged into LDS via async global->LDS b128 copies (ASYNCcnt),
// A fragments read as ds_load_b64 (bank-conflict-free via LDSPAD),
// B read direct (64KB weight matrix, cache-resident, lane-contiguous).
// Per k-step/wave: 1 ds_load_b64 + 16 coalesced b32 + 8x V_WMMA_F32_16X16X4_F32.
// Full blocks (781/782) take a guard-free store path.
// =====================================================================
__global__ __launch_bounds__(256) void k_gemm_node_wmma(const float* __restrict__ A,
                                                        const float* __restrict__ B,
                                                        float* __restrict__ C, int M) {
    __shared__ float As[128 * LDSPAD];
    const int t = threadIdx.x;
    const int rbase = blockIdx.x * 128;

    // ---- stage A tile (128 rows x 128 cols) into LDS, coalesced 16B/lane ----
    {
        unsigned lbase = (unsigned)(size_t)(&As[0]);
        for (int i = 0; i < 16; ++i) {
            int c = t + i * 256;                    // chunk 0..4095, 16B each
            int row = c >> 5, q = c & 31;           // 32 chunks per 512B row
            int grow = rbase + row; grow = grow < M ? grow : (M - 1);
            unsigned long long ga = (unsigned long long)(A + (long)grow * FIN + q * 4);
            unsigned la = lbase + (unsigned)(row * (LDSPAD * 4) + q * 16);
            asm volatile("global_load_async_to_lds_b128 %0, %1, off"
                         :: "v"(la), "v"(ga) : "memory");
        }
        asm volatile("s_wait_asynccnt 0" ::: "memory");
        __syncthreads();
    }

    const int lane = t & 31;
    const int w    = t >> 5;
    const int half = lane >> 4;
    const int lr   = lane & 15;
    const int rowoff = (w * 16 + lr) * LDSPAD;

    v8f acc[8];
#pragma unroll
    for (int n = 0; n < 8; ++n) acc[n] = (v8f){};

    for (int k0 = 0; k0 < 128; k0 += 4) {
        int ka = k0 + 2 * half;
        v2f a;
        a.x = As[rowoff + ka];
        a.y = As[rowoff + ka + 1];
#pragma unroll
        for (int n = 0; n < 8; ++n) {
            v2f b;
            b.x = B[ka * FIN + n * 16 + lr];
            b.y = B[(ka + 1) * FIN + n * 16 + lr];
            acc[n] = __builtin_amdgcn_wmma_f32_16x16x4_f32(false, a, false, b, (short)0,
                                                           acc[n], false, false);
        }
    }

    const int rw = rbase + w * 16;
    if (rbase + 128 <= M) {
        // uniform fast path: every row valid, no guards
        float* Cw = C + (long)(rw + 8 * half) * FIN + lr;
#pragma unroll
        for (int v = 0; v < 8; ++v) {
            float* Cr = Cw + (long)v * FIN;
#pragma unroll
            for (int n = 0; n < 8; ++n) Cr[n * 16] = acc[n][v];
        }
    } else {
#pragma unroll
        for (int v = 0; v < 8; ++v) {
            int m = rw + v + 8 * half;
            if (m < M) {
#pragma unroll
                for (int n = 0; n < 8; ++n) C[(long)m * FIN + n * 16 + lr] = acc[n][v];
            }
        }
    }
}

// ---------------- generic fp32 WMMA GEMM (small readout matrices) ----------------
// branchless guards: clamped addresses + v_cndmask zeroing (no exec-mask branches)
__global__ void k_gemm_wmma(const float* __restrict__ A, int lda,
                            const float* __restrict__ B, int ldb,
                            float* __restrict__ C, int ldc,
                            int M, int Kv, const float* __restrict__ bias, int relu) {
    int lane = threadIdx.x & 31;
    int half = lane >> 4;
    int lr   = lane & 15;
    int r0 = blockIdx.x * 16;
    int c0 = blockIdx.y * 16;
    int arowr = r0 + lr;
    int arow = arowr < M ? arowr : (M - 1);
    bool arv = arowr < M;
    v8f acc = {};
    for (int k0 = 0; k0 < Kv; k0 += 4) {
        int ka = k0 + 2 * half;
        int k1 = ka < Kv - 1 ? ka : (Kv - 1);
        int k2 = ka + 1 < Kv - 1 ? ka + 1 : (Kv - 1);
        float ax = A[(long)arow * lda + k1];
        float ay = A[(long)arow * lda + k2];
        float bx = B[(long)k1 * ldb + c0 + lr];
        float by = B[(long)k2 * ldb + c0 + lr];
        v2f a, b;
        a.x = (arv && ka < Kv)     ? ax : 0.f;
        a.y = (arv && ka + 1 < Kv) ? ay : 0.f;
        b.x = (ka < Kv)     ? bx : 0.f;
        b.y = (ka + 1 < Kv) ? by : 0.f;
        acc = __builtin_amdgcn_wmma_f32_16x16x4_f32(false, a, false, b, (short)0, acc, false, false);
    }
    float bv = bias ? bias[c0 + lr] : 0.f;
#pragma unroll
    for (int v = 0; v < 8; v++) {
        int m = r0 + v + 8 * half;
        if (m < M) {
            float val = acc[v] + bv;
            if (relu) val = fmaxf(val, 0.f);
            C[(long)m * ldc + c0 + lr] = val;
        }
    }
}

// ---------------- per-node attention logits: alpha_s / alpha_d ----------------
__global__ void k_alpha_nodes(const float* __restrict__ hp, const float* __restrict__ as_,
                              const float* __restrict__ ad_, float* __restrict__ als,
                              float* __restrict__ ald) {
    long t = blockIdx.x * (long)blockDim.x + threadIdx.x;
    if (t >= (long)Nn * HEADS) return;
    long i = t >> 1; int h = (int)(t & 1);
    const float* row = hp + i * (HEADS * HIDd) + h * HIDd;
    float s = 0.f, d2 = 0.f;
    for (int j = 0; j < HIDd; j++) { float v = row[j]; s += v * as_[h * HIDd + j]; d2 += v * ad_[h * HIDd + j]; }
    als[t] = s; ald[t] = d2;
}

// ---------------- per-edge logits + leaky_relu + segment max ----------------
__global__ void k_edge_alpha(const int* __restrict__ ei, const float* __restrict__ ea,
                             const float* __restrict__ eal, const float* __restrict__ ve,
                             const float* __restrict__ als, const float* __restrict__ ald,
                             float* __restrict__ alpha, unsigned* __restrict__ mkey) {
    long e = blockIdx.x * (long)blockDim.x + threadIdx.x;
    if (e >= (long)EfE) return;
    int src, dst; const float* eap;
    if (e < Ee) { src = ei[e]; dst = ei[Ee + e]; eap = ea + e * EDGED; }
    else        { src = dst = (int)(e - Ee);     eap = eal + (long)dst * EDGED; }
    float a0 = 0.f, a1 = 0.f;
    for (int d = 0; d < EDGED; d++) { float v = eap[d]; a0 += v * ve[d * 2]; a1 += v * ve[d * 2 + 1]; }
    float t0 = als[(long)src * 2]     + ald[(long)dst * 2]     + a0;
    float t1 = als[(long)src * 2 + 1] + ald[(long)dst * 2 + 1] + a1;
    t0 = t0 > 0.f ? t0 : NSLOPE * t0;
    t1 = t1 > 0.f ? t1 : NSLOPE * t1;
    alpha[e * 2] = t0; alpha[e * 2 + 1] = t1;
    atomicMax(&mkey[(long)dst * 2],     enc_f(t0));
    atomicMax(&mkey[(long)dst * 2 + 1], enc_f(t1));
}

// ---------------- exp(alpha - max) and denominator ----------------
__global__ void k_edge_exp(const int* __restrict__ ei, const unsigned* __restrict__ mkey,
                           float* __restrict__ alpha, float* __restrict__ den) {
    long t = blockIdx.x * (long)blockDim.x + threadIdx.x;
    if (t >= (long)EfE * 2) return;
    long e = t >> 1; int h = (int)(t & 1);
    int dst = (e < Ee) ? ei[Ee + e] : (int)(e - Ee);
    float ex = __expf(alpha[t] - dec_f(mkey[(long)dst * 2 + h]));
    alpha[t] = ex;
    atomicAdd(&den[(long)dst * 2 + h], ex);
}

// ---------------- out init with bias ----------------
__global__ void k_bias_init(float* __restrict__ out, const float* __restrict__ bias) {
    long t = blockIdx.x * (long)blockDim.x + threadIdx.x;
    if (t >= (long)Nn * FIN) return;
    out[t] = bias[t & 127];
}

// ---------------- weighted message scatter: one block per edge ----------------
__global__ void k_msg(const int* __restrict__ ei, const float* __restrict__ hp,
                      const float* __restrict__ alpha, const float* __restrict__ den,
                      float* __restrict__ out) {
    long e = blockIdx.x;
    int j = threadIdx.x;          // 0..127
    int h = j >> 6;
    int src, dst;
    if (e < Ee) { src = ei[e]; dst = ei[Ee + e]; } else { src = dst = (int)(e - Ee); }
    float w = alpha[e * 2 + h] / den[(long)dst * 2 + h];
    atomicAdd(&out[(long)dst * FIN + j], hp[(long)src * FIN + j] * w);
}

__global__ void k_relu(float* __restrict__ p, long n) {
    long i = blockIdx.x * (long)blockDim.x + threadIdx.x;
    if (i < n) p[i] = fmaxf(p[i], 0.f);
}

// ---------------- z = concat(h[node0], dtype_vec) ----------------
__global__ void k_build_z(const float* __restrict__ h, const int* __restrict__ node0,
                          const float* __restrict__ dtv, float* __restrict__ z) {
    int t = blockIdx.x * blockDim.x + threadIdx.x;
    if (t >= BG * SHIN) return;
    int g = t / SHIN, c = t % SHIN;
    z[t] = (c < FIN) ? h[(long)node0[g] * FIN + c] : dtv[g * DTS + (c - FIN)];
}

// ---------------- final per-task projection (64 -> HEAD_OUT[k]) ----------------
__global__ void k_head2(const float* __restrict__ hid, const float* __restrict__ W2,
                        const float* __restrict__ b2, float* __restrict__ out) {
    const int HO[7]  = {3, 3, 3, 9, 1, 1, 8};
    const int OFF[7] = {0, 3000, 6000, 9000, 18000, 19000, 20000};
    int t = blockIdx.x * blockDim.x + threadIdx.x;
    if (t >= 7 * BG * MAXO) return;
    int k = t / (BG * MAXO); int r = t % (BG * MAXO); int g = r / MAXO, o = r % MAXO;
    if (o >= HO[k]) return;
    const float* hrow = hid + ((long)k * BG + g) * TASKH;
    const float* w = W2 + (long)k * TASKH * MAXO;
    float s = b2[k * MAXO + o];
    for (int j = 0; j < TASKH; j++) s += hrow[j] * w[j * MAXO + o];
    out[OFF[k] + g * HO[k] + o] = s;
}

static inline dim3 g1(long n, int b) { return dim3((unsigned)((n + b - 1) / b)); }

extern "C" void kernel_launch(void* const* d_in, const int* in_sizes, int n_in,
                              void* d_out, int out_size, void* d_ws, size_t ws_size,
                              hipStream_t stream) {
    const float* x       = (const float*)d_in[0];
    const int*   ei      = (const int*)  d_in[1];
    const int*   batch   = (const int*)  d_in[2];
    const float* eattr   = (const float*)d_in[3];
    const float* gat_W   = (const float*)d_in[4];
    const float* att_src = (const float*)d_in[5];
    const float* att_dst = (const float*)d_in[6];
    const float* edge_W  = (const float*)d_in[7];
    const float* att_e   = (const float*)d_in[8];
    const float* gat_b   = (const float*)d_in[9];
    const float* sh_W    = (const float*)d_in[10];
    const float* sh_b    = (const float*)d_in[11];
    const float* h_W1    = (const float*)d_in[12];
    const float* h_b1    = (const float*)d_in[13];
    const float* h_W2    = (const float*)d_in[14];
    const float* h_b2    = (const float*)d_in[15];
    float* out = (float*)d_out;
    (void)in_sizes; (void)n_in; (void)out_size; (void)ws_size;

    // workspace layout (fp32 words)
    float* w = (float*)d_ws;
    float*    deg   = w;                 w += Nn;
    float*    eal   = w;                 w += (long)Nn * EDGED;
    float*    hp    = w;                 w += (long)Nn * FIN;
    float*    buf0  = w;                 w += (long)Nn * FIN;
    float*    buf1  = w;                 w += (long)Nn * FIN;
    float*    als   = w;                 w += (long)Nn * HEADS;
    float*    ald   = w;                 w += (long)Nn * HEADS;
    float*    alpha = w;                 w += (long)EfE * HEADS;
    unsigned* mkey  = (unsigned*)w;      w += (long)Nn * HEADS;
    float*    den   = w;                 w += (long)Nn * HEADS;
    float*    ve    = w;                 w += EDGED * HEADS;
    int*      node0 = (int*)w;           w += BG;
    float*    dtv   = w;                 w += BG * DTS;
    float*    z     = w;                 w += BG * SHIN;
    float*    zs    = w;                 w += BG * HIDd;
    float*    hid   = w;                 w += (long)7 * BG * TASKH;

    // degree + mean self-loop edge features
    k_zero_f<<<g1(Nn, 256), 256, 0, stream>>>(deg, Nn);
    k_zero_f<<<g1((long)Nn * EDGED, 256), 256, 0, stream>>>(eal, (long)Nn * EDGED);
    k_deg_ealoop<<<g1((long)Ee * EDGED, 256), 256, 0, stream>>>(ei, eattr, deg, eal);
    k_ealoop_div<<<g1((long)Nn * EDGED, 256), 256, 0, stream>>>(eal, deg);
    k_node0<<<g1(BG, 256), 256, 0, stream>>>(batch, x, node0, dtv);

    const float* hin = x;
    float* bufs[2] = {buf0, buf1};
    float* hout = nullptr;
    for (int l = 0; l < NL; l++) {
        hout = bufs[l & 1];
        k_ve<<<1, 64, 0, stream>>>(edge_W + (long)l * EDGED * FIN, att_e + (long)l * HEADS * HIDd, ve);
        // hp = h @ W[l]   (fp32 WMMA, async LDS-staged A)
        k_gemm_node_wmma<<<dim3((Nn + 127) / 128), 256, 0, stream>>>(
            hin, gat_W + (long)l * FIN * FIN, hp, Nn);
        k_alpha_nodes<<<g1((long)Nn * HEADS, 256), 256, 0, stream>>>(
            hp, att_src + (long)l * HEADS * HIDd, att_dst + (long)l * HEADS * HIDd, als, ald);
        k_zero_f<<<g1((long)Nn * HEADS, 256), 256, 0, stream>>>((float*)mkey, (long)Nn * HEADS);
        k_zero_f<<<g1((long)Nn * HEADS, 256), 256, 0, stream>>>(den, (long)Nn * HEADS);
        k_edge_alpha<<<g1(EfE, 256), 256, 0, stream>>>(ei, eattr, eal, ve, als, ald, alpha, mkey);
        k_edge_exp<<<g1((long)EfE * 2, 256), 256, 0, stream>>>(ei, mkey, alpha, den);
        k_bias_init<<<g1((long)Nn * FIN, 256), 256, 0, stream>>>(hout, gat_b + (long)l * FIN);
        k_msg<<<dim3(EfE), 128, 0, stream>>>(ei, hp, alpha, den, hout);
        if (l < NL - 1)
            k_relu<<<g1((long)Nn * FIN, 256), 256, 0, stream>>>(hout, (long)Nn * FIN);
        hin = hout;
    }

    // readout MLP
    k_build_z<<<g1(BG * SHIN, 256), 256, 0, stream>>>(hout, node0, dtv, z);
    k_gemm_wmma<<<dim3((BG + 15) / 16, HIDd / 16), 32, 0, stream>>>(
        z, SHIN, sh_W, HIDd, zs, HIDd, BG, SHIN, sh_b, 1);
    for (int k = 0; k < 7; k++) {
        k_gemm_wmma<<<dim3((BG + 15) / 16, TASKH / 16), 32, 0, stream>>>(
            zs, HIDd, h_W1 + (long)k * HIDd * TASKH, TASKH,
            hid + (long)k * BG * TASKH, TASKH, BG, HIDd, h_b1 + (long)k * TASKH, 1);
    }
    k_head2<<<g1(7 * BG * MAXO, 256), 256, 0, stream>>>(hid, h_W2, h_b2, out);
}